// Proposal_52836687675621
// MI455X (gfx1250) — compile-verified
//
#include <hip/hip_runtime.h>
#include <hip/hip_bf16.h>

#define NN    20000
#define SORTN 32768
#define PRE   6000
#define POST  300
#define MW    188          // ceil(6000/32)
#define NMS_T 0.7f

typedef unsigned int       u32;
typedef unsigned long long u64;
typedef float v2f  __attribute__((ext_vector_type(2)));
typedef float v8f  __attribute__((ext_vector_type(8)));
typedef u32   u32x4v __attribute__((ext_vector_type(4)));
typedef u32   u32x8v __attribute__((ext_vector_type(8)));

#if defined(__HIP_DEVICE_COMPILE__) && __has_builtin(__builtin_amdgcn_wmma_f32_16x16x4_f32)
#define HAVE_WMMA_F32X4 1
#endif

__device__ __forceinline__ u32 sortable_f32(float s) {
  u32 b = __float_as_uint(s);
  return (b & 0x80000000u) ? ~b : (b | 0x80000000u);
}

// min/max for known non-negative floats via unsigned compare: avoids the
// v_max_num x,x canonicalize ops clang inserts for minnum/maxnum on loads.
__device__ __forceinline__ float fmin_nn(float a, float b) {
  u32 ua = __float_as_uint(a), ub = __float_as_uint(b);
  return __uint_as_float(ua < ub ? ua : ub);
}
__device__ __forceinline__ float fmax_nn(float a, float b) {
  u32 ua = __float_as_uint(a), ub = __float_as_uint(b);
  return __uint_as_float(ua > ub ? ua : ub);
}

// ------- 1) fused: decode boxes + build keys + LDS bitonic phases k=2..512 ---
__global__ void __launch_bounds__(256)
decode_sort_local(const float* __restrict__ loc,
                  const float* __restrict__ fg,
                  const float* __restrict__ anc,
                  const int* __restrict__ scale,
                  const int* __restrict__ imw,
                  const int* __restrict__ imh,
                  float4* __restrict__ boxes,
                  u64* __restrict__ keys) {
  __shared__ u64 sk[512];
  const int t = threadIdx.x;
  const int base = blockIdx.x * 512;
  float W  = (float)imw[0];
  float H  = (float)imh[0];
  float ms = 16.0f * (float)scale[0];
#pragma unroll
  for (int e0 = 0; e0 < 2; ++e0) {
    int i = base + t + e0 * 256;
    u64 key = 0ull;
    if (i < NN) {
      float ax = anc[i*4+0], ay = anc[i*4+1], aw = anc[i*4+2], ah = anc[i*4+3];
      float l0 = loc[i*4+0], l1 = loc[i*4+1], l2 = loc[i*4+2], l3 = loc[i*4+3];
      float cx = l0 * aw * 0.1f + ax;
      float cy = l1 * ah * 0.1f + ay;
      float bw = expf(l2 * 0.2f) * aw;
      float bh = expf(l3 * 0.2f) * ah;
      float x1 = fminf(fmaxf(cx - bw * 0.5f, 0.0f), W);
      float y1 = fminf(fmaxf(cy - bh * 0.5f, 0.0f), H);
      float x2 = fminf(fmaxf(cx + bw * 0.5f, 0.0f), W);
      float y2 = fminf(fmaxf(cy + bh * 0.5f, 0.0f), H);
      bool valid = ((y2 - y1) >= ms) && ((x2 - x1) >= ms);
      float score = valid ? fg[i] : -__builtin_inff();
      boxes[i] = make_float4(x1, y1, x2, y2);
      key = ((u64)sortable_f32(score) << 32) | (u32)(0xFFFFFFFFu - (u32)i);
    }
    sk[t + e0 * 256] = key;
  }
  __syncthreads();
  for (int k = 2; k <= 512; k <<= 1) {
    for (int j = k >> 1; j > 0; j >>= 1) {
#pragma unroll
      for (int e0 = 0; e0 < 2; ++e0) {
        int e = t + e0 * 256;
        int ixj = e ^ j;
        if (ixj > e) {
          u64 a = sk[e], b = sk[ixj];
          bool up = (((base + e) & k) == 0);
          if (up ? (a < b) : (a > b)) { sk[e] = b; sk[ixj] = a; }
        }
      }
      __syncthreads();
    }
  }
  keys[base + t]       = sk[t];
  keys[base + t + 256] = sk[t + 256];
}

// ---------------- 2) bitonic sort (descending), launch-fused -----------------
// Global compare-exchange: used only for strides j >= 512.
__global__ void bitonic_step(u64* __restrict__ keys, int k, int j) {
  int i = blockIdx.x * blockDim.x + threadIdx.x;
  int ixj = i ^ j;
  if (ixj <= i) return;
  u64 a = keys[i], b = keys[ixj];
  bool up = ((i & k) == 0);
  if (up ? (a < b) : (a > b)) { keys[i] = b; keys[ixj] = a; }
}

// LDS-resident tail of phase k: strides j = 256..1
__global__ void __launch_bounds__(256) bitonic_local_tail(u64* __restrict__ keys, int k) {
  __shared__ u64 sk[512];
  const int t = threadIdx.x;
  const int base = blockIdx.x * 512;
  sk[t]       = keys[base + t];
  sk[t + 256] = keys[base + t + 256];
  __syncthreads();
  for (int j = 256; j > 0; j >>= 1) {
#pragma unroll
    for (int e0 = 0; e0 < 2; ++e0) {
      int e = t + e0 * 256;
      int ixj = e ^ j;
      if (ixj > e) {
        u64 a = sk[e], b = sk[ixj];
        bool up = (((base + e) & k) == 0);
        if (up ? (a < b) : (a > b)) { sk[e] = b; sk[ixj] = a; }
      }
    }
    __syncthreads();
  }
  keys[base + t]       = sk[t];
  keys[base + t + 256] = sk[t + 256];
}

// ---------------- 3) gather top-PRE boxes in score order ---------------------
__global__ void gather_kernel(const u64* __restrict__ keys,
                              const float4* __restrict__ boxes,
                              float4* __restrict__ sb) {
  int i = blockIdx.x * blockDim.x + threadIdx.x;
  if (i >= PRE) return;
  u32 idx = 0xFFFFFFFFu - (u32)(keys[i] & 0xFFFFFFFFull);
  sb[i] = boxes[idx];
}

// ---------------- 4) pairwise IoU -> suppression bitmask ---------------------
// block: 128 thr (4 waves). rows i0..i0+15, cols j0..j0+127 (one 32-col word/wave).
// Ballot predicate is ONLY the float IoU test; the j>i diagonal constraint and
// the partial-last-word constraint are applied as write-time bitmask ANDs.
__global__ void __launch_bounds__(128)
mask_kernel(const float4* __restrict__ sb, u32* __restrict__ mask) {
  __shared__ float4 colBox[128];
  __shared__ float4 rowBox[16];
  const int i0 = blockIdx.y * 16;
  const int j0 = blockIdx.x * 128;
  const int wave = threadIdx.x >> 5;
  const int lane = threadIdx.x & 31;
  const u32 wordIdx = (u32)(j0 / 32 + wave);

  // Entire tile strictly below the diagonal -> all bits zero, skip the math.
  if (j0 + 128 <= i0) {
    if (lane < 16) mask[(u32)(i0 + lane) * MW + wordIdx] = 0u;
    return;
  }

  int cnt = PRE - j0; if (cnt > 128) cnt = 128;
  if (threadIdx.x < 16) rowBox[threadIdx.x] = sb[i0 + threadIdx.x];

  // Stage the column tile into LDS with the Tensor Data Mover (async DMA).
  if (threadIdx.x < 32) {                      // one wave issues; TDM ignores EXEC
    unsigned long long ga = (unsigned long long)(const void*)(sb + j0);
    u32 lds   = (u32)(unsigned long long)(const void*)&colBox[0];
    u32 elems = (u32)cnt * 4u;                 // f32 elements in the 1-D tile
    u32x4v g0; u32x8v g1;
    g0[0] = 1u;                                         // count=1, user descriptor
    g0[1] = lds;                                        // lds_addr
    g0[2] = (u32)ga;                                    // global_addr[31:0]
    g0[3] = (u32)((ga >> 32) & 0x01FFFFFFull)           // global_addr[56:32]
          | (2u << 30);                                 // type = 2 (image)
    g1[0] = (2u << 16);                                 // data_size = 4B
    g1[1] = (elems & 0xFFFFu) << 16;                    // tensor_dim0[15:0]
    g1[2] = (elems >> 16) | (1u << 16);                 // tensor_dim0[31:16], dim1=1
    g1[3] = (elems & 0xFFFFu) << 16;                    // tile_dim0
    g1[4] = 0u;                                         // tile_dim1/2 unused
    g1[5] = elems;                                      // tensor_dim0_stride[31:0]
    g1[6] = 0u;
    g1[7] = 0u;
    asm volatile("tensor_load_to_lds %0, %1" :: "s"(g0), "s"(g1) : "memory");
  }
  __builtin_amdgcn_s_wait_tensorcnt(0);
  __syncthreads();

  const int n   = lane & 15;
  const bool lo = (lane < 16);

  float4 rbn  = rowBox[n];
  float areaI = (rbn.z - rbn.x) * (rbn.w - rbn.y);
  v2f A; A.x = lo ? areaI : 0.0f; A.y = lo ? 1.0f : 0.0f;   // A(16x4): [area_i, 1, 0, 0]

  // Hoist this lane's 8 row boxes out of both subtile loops.
  float4 ib[8];
#pragma unroll
  for (int r = 0; r < 8; ++r) ib[r] = rowBox[r + (lo ? 0 : 8)];

  u32 b0[8], b1[8];
#pragma unroll
  for (int t = 0; t < 2; ++t) {
    float4 cbx  = colBox[wave * 32 + t * 16 + n];
    float areaJ = (cbx.z - cbx.x) * (cbx.w - cbx.y);
    v2f B; B.x = lo ? 1.0f : 0.0f; B.y = lo ? areaJ : 0.0f; // B(4x16): [1; area_j; 0; 0]
    v8f C = {};
#if HAVE_WMMA_F32X4
    // C[M][N] = area_i[M] + area_j[N] on the matrix pipe
    C = __builtin_amdgcn_wmma_f32_16x16x4_f32(false, A, false, B,
                                              (short)0, C, false, false);
#endif
#pragma unroll
    for (int r = 0; r < 8; ++r) {
      // coords are clipped to [0, W/H]: unsigned-int compare == float compare
      float iw = fmin_nn(ib[r].z, cbx.z) - fmax_nn(ib[r].x, cbx.x);
      float ih = fmin_nn(ib[r].w, cbx.w) - fmax_nn(ib[r].y, cbx.y);
      iw = fmaxf(iw, 0.0f); ih = fmaxf(ih, 0.0f);
      float inter = iw * ih;
#if HAVE_WMMA_F32X4
      float aSum = C[r];
#else
      float aSum = (ib[r].z - ib[r].x) * (ib[r].w - ib[r].y) + areaJ;
#endif
      // iou > T  <=>  inter > T * union   (union > 0), division-free
      u32 bal = __builtin_amdgcn_ballot_w32(inter > NMS_T * (aSum - inter + 1e-12f));
      if (t == 0) b0[r] = bal; else b1[r] = bal;
    }
  }
  // ballot r: bits[15:0] = row r, bits[31:16] = row r+8 (WMMA C layout)
  if (lane < 16) {
    int r = lane & 7;
    u32 w = (lane < 8) ? ((b0[r] & 0xFFFFu) | ((b1[r] & 0xFFFFu) << 16))
                       : ((b0[r] >> 16)     | (b1[r] & 0xFFFF0000u));
    const int gi = i0 + lane;
    const int d  = gi - (int)wordIdx * 32;
    u32 diag = (d < 0) ? 0xFFFFFFFFu : ((d >= 31) ? 0u : (0xFFFFFFFFu << (d + 1)));
    u32 colv = (wordIdx == MW - 1) ? ((1u << (PRE & 31)) - 1u) : 0xFFFFFFFFu;
    mask[(u32)gi * MW + wordIdx] = w & diag & colv;
  }
}

// ---------------- 5) serial greedy NMS (bit-scan) + emit first POST ----------
__global__ void nms_reduce(const u32* __restrict__ mask,
                           const float4* __restrict__ sb,
                           const int* __restrict__ scale,
                           float* __restrict__ out) {
  __shared__ u32 supp[MW];
  const int lane = threadIdx.x;         // 32 threads, single wave
  float ms = 16.0f * (float)scale[0];
  for (int w = lane; w < MW; w += 32) {
    u32 v = 0u;
    for (int b = 0; b < 32; ++b) {
      int j = w * 32 + b;
      bool dead = true;
      if (j < PRE) {
        float4 bx = sb[j];
        dead = !(((bx.w - bx.y) >= ms) && ((bx.z - bx.x) >= ms));
      }
      if (dead) v |= (1u << b);
    }
    supp[w] = v;
  }
  __syncthreads();
  volatile u32* vs = supp;
  for (int w = 0; w < MW; ++w) {
    u32 alive = ~vs[w];
    while (alive) {
      int b = __builtin_ctz(alive);
      int i = w * 32 + b;
      if (i >= PRE) break;
      // prefetch the next candidate's mask row while we OR this one
      u32 nxt = alive & (alive - 1u);
      if (nxt) {
        int b2 = __builtin_ctz(nxt);
        const u32* nrow = mask + (size_t)(w * 32 + b2) * MW;
        if (lane < 6) __builtin_prefetch((const void*)(nrow + lane * 32), 0, 0);
      }
      const u32* row = mask + (size_t)i * MW;
      for (int q = lane; q < MW; q += 32) vs[q] |= row[q];
      alive = ~vs[w] & (0xFFFFFFFEu << b);   // bits strictly above b still alive
    }
  }
  __syncthreads();
  for (int k = lane; k < POST * 4; k += 32) out[k] = 0.0f;
  if (lane == 0) {
    int cnt = 0;
    for (int w = 0; w < MW && cnt < POST; ++w) {
      u32 alive = ~vs[w];
      while (alive && cnt < POST) {
        int b = __builtin_ctz(alive);
        int i = w * 32 + b;
        if (i >= PRE) break;
        float4 bx = sb[i];
        out[cnt * 4 + 0] = bx.x; out[cnt * 4 + 1] = bx.y;
        out[cnt * 4 + 2] = bx.z; out[cnt * 4 + 3] = bx.w;
        ++cnt;
        alive &= alive - 1u;
      }
    }
  }
}

// ---------------- host launcher ---------------------------------------------
extern "C" void kernel_launch(void* const* d_in, const int* in_sizes, int n_in,
                              void* d_out, int out_size, void* d_ws, size_t ws_size,
                              hipStream_t stream) {
  const float* loc  = (const float*)d_in[0];   // [1,N,4]
  const float* fg   = (const float*)d_in[1];   // [1,N]
  const float* anc  = (const float*)d_in[2];   // [N,4]
  const int*  scale = (const int*)d_in[3];
  const int*  imw   = (const int*)d_in[4];
  const int*  imh   = (const int*)d_in[5];
  float* out = (float*)d_out;

  char* ws = (char*)d_ws;
  float4* boxes = (float4*)(ws + 0);               // 320,000 B
  u64*    keys  = (u64*)  (ws + 320000);           // 262,144 B
  float4* sb    = (float4*)(ws + 582144);          //  96,000 B
  u32*    mask  = (u32*)  (ws + 678144);           // 4,512,000 B

  // decode fused with the first LDS-resident sort phases (k=2..512)
  decode_sort_local<<<SORTN / 512, 256, 0, stream>>>(loc, fg, anc, scale, imw, imh,
                                                     boxes, keys);
  for (int k = 1024; k <= SORTN; k <<= 1) {
    for (int j = k >> 1; j >= 512; j >>= 1)
      bitonic_step<<<SORTN / 256, 256, 0, stream>>>(keys, k, j);
    bitonic_local_tail<<<SORTN / 512, 256, 0, stream>>>(keys, k);
  }
  gather_kernel<<<(PRE + 255) / 256, 256, 0, stream>>>(keys, boxes, sb);
  mask_kernel<<<dim3((PRE + 127) / 128, PRE / 16), 128, 0, stream>>>(sb, mask);
  nms_reduce<<<1, 32, 0, stream>>>(mask, sb, scale, out);
}